// MambaModule_18253611008834
// MI455X (gfx1250) — compile-verified
//
#include <hip/hip_runtime.h>

// ---------------------------------------------------------------------------
// MI455X (gfx1250) implementation of the MambaModule forward pass.
//
// Precision/throughput strategy: ~30 GFLOP fp32 vs ~0.3-0.5 GB of traffic ->
// memory/latency bound at 23.3 TB/s, so keep full fp32 everywhere (bit-class
// identical to the reference) and put ALL dense math (Mamba GEMMs *and* every
// convolution, via implicit GEMM) on the native fp32 matrix op
// V_WMMA_F32_16X16X4_F32 (1024 fp32 FMA per wave-issue vs 32 on the VALU).
// ---------------------------------------------------------------------------

typedef __attribute__((ext_vector_type(2))) float v2f;
typedef __attribute__((ext_vector_type(8))) float v8f;

static __device__ __forceinline__ float siluf(float x) {
  return x / (1.f + expf(-x));
}

// ---------------------------------------------------------------------------
// BatchNorm statistics: one block per channel, produce fused scale/shift.
// scale = g * rsqrt(var + eps); shift = b - mean * scale
// ---------------------------------------------------------------------------
__global__ void bn_stats_kernel(const float* __restrict__ x, int C, int HW, int NB,
                                const float* __restrict__ g, const float* __restrict__ b,
                                float* __restrict__ ss) {
  int c = blockIdx.x;
  int tid = threadIdx.x;
  int cnt = NB * HW;
  float s1 = 0.f, s2 = 0.f;
  for (int i = tid; i < cnt; i += 256) {
    int bb = i / HW, p = i - bb * HW;
    float v = x[((size_t)bb * C + c) * HW + p];
    s1 += v; s2 += v * v;
  }
  __shared__ float sh1[256], sh2[256];
  sh1[tid] = s1; sh2[tid] = s2; __syncthreads();
  for (int off = 128; off > 0; off >>= 1) {
    if (tid < off) { sh1[tid] += sh1[tid + off]; sh2[tid] += sh2[tid + off]; }
    __syncthreads();
  }
  if (tid == 0) {
    float inv = 1.f / (float)cnt;
    float mean = sh1[0] * inv;
    float var  = sh2[0] * inv - mean * mean;
    float sc = g[c] * rsqrtf(var + 1e-5f);
    ss[2 * c]     = sc;
    ss[2 * c + 1] = b[c] - mean * sc;
  }
}

// Stats over the *virtual* input f = concat(high, features + tile(high,4)).
__global__ void bn_in_stats_kernel(const float* __restrict__ feat,
                                   const float* __restrict__ high,
                                   const float* __restrict__ g,
                                   const float* __restrict__ b,
                                   float* __restrict__ ss) {
  int c = blockIdx.x;           // 0..239
  int tid = threadIdx.x;
  float s1 = 0.f, s2 = 0.f;
  for (int i = tid; i < 10 * 4096; i += 256) {
    int bb = i >> 12, p = i & 4095;
    float v;
    if (c < 48) v = high[((size_t)bb * 48 + c) * 4096 + p];
    else {
      int cc = c - 48;
      v = feat[((size_t)bb * 192 + cc) * 4096 + p] +
          high[((size_t)bb * 48 + (cc % 48)) * 4096 + p];
    }
    s1 += v; s2 += v * v;
  }
  __shared__ float sh1[256], sh2[256];
  sh1[tid] = s1; sh2[tid] = s2; __syncthreads();
  for (int off = 128; off > 0; off >>= 1) {
    if (tid < off) { sh1[tid] += sh1[tid + off]; sh2[tid] += sh2[tid + off]; }
    __syncthreads();
  }
  if (tid == 0) {
    float inv = 1.f / 40960.f;
    float mean = sh1[0] * inv;
    float var  = sh2[0] * inv - mean * mean;
    float sc = g[c] * rsqrtf(var + 1e-5f);
    ss[2 * c]     = sc;
    ss[2 * c + 1] = b[c] - mean * sc;
  }
}

// Materialize f = BN(concat(high, features + tile(high,4)))  -> (10,240,64,64)
__global__ void build_f_kernel(const float* __restrict__ feat,
                               const float* __restrict__ high,
                               const float* __restrict__ ss,
                               float* __restrict__ f) {
  int idx = blockIdx.x * 256 + threadIdx.x;   // total 9,830,400 (exact multiple)
  int p = idx & 4095; int t = idx >> 12;
  int c = t % 240; int b = t / 240;
  float v;
  if (c < 48) v = high[((size_t)b * 48 + c) * 4096 + p];
  else {
    int cc = c - 48;
    v = feat[((size_t)b * 192 + cc) * 4096 + p] +
        high[((size_t)b * 48 + (cc % 48)) * 4096 + p];
  }
  f[idx] = v * ss[2 * c] + ss[2 * c + 1];
}

// x = silu(x*scale + shift), NCHW in place
__global__ void bn_apply_silu_kernel(float* __restrict__ x, const float* __restrict__ ss,
                                     int C, int HW, int total) {
  int idx = blockIdx.x * 256 + threadIdx.x;
  if (idx >= total) return;
  int c = (idx / HW) % C;
  float v = x[idx] * ss[2 * c] + ss[2 * c + 1];
  x[idx] = siluf(v);
}

// ---------------------------------------------------------------------------
// WMMA implicit-GEMM convolution (pad = KS/2, square KS in {1,3}, grouped).
// One wave computes a tile of 16 output channels x 16 output pixels (along W).
//   A[m][k] = w[ocTile+m][ic=k][tap]        (16xK weight slab)
//   B[k][n] = in[icb+k][ih][iw(n)]          (Kx16 input slab, 0 for OOB cols)
// K steps by 4 per V_WMMA_F32_16X16X4_F32. Column OOB handled branchlessly
// (clamped address + select 0) so EXEC is all-1s at every WMMA; row OOB taps
// are skipped wave-uniformly. Fragment layouts per ISA 7.12.2.
// EPI: 0=none, 1=+bias, 2=+bias then silu
// ---------------------------------------------------------------------------
template <int KS, int EPI>
__global__ void __launch_bounds__(128)
conv_wmma_kernel(const float* __restrict__ in, const float* __restrict__ w,
                 const float* __restrict__ bias, float* __restrict__ out,
                 int B, int Cin, int Cout, int Hin, int Win, int Hout, int Wout,
                 int stride, int icPerG, int ocPerG, int icOffset) {
  const int lane  = threadIdx.x & 31;
  const int rowLo = lane & 15;
  const int kSel  = (lane >> 4) << 1;       // 0 for lanes 0-15, 2 for 16-31
  int waveId = blockIdx.x * 4 + (threadIdx.x >> 5);
  const int nW = Wout >> 4;
  const int wt = waveId % nW; waveId /= nW;
  const int ho = waveId % Hout; waveId /= Hout;
  const int nO = Cout >> 4;
  const int ot = waveId % nO;
  const int b  = waveId / nO;
  const int g   = (ot << 4) / ocPerG;       // ocPerG is a multiple of 16
  const int icb = g * icPerG + icOffset;
  const int pad = KS >> 1;
  const int KS2 = KS * KS;
  const size_t chStride = (size_t)Hin * Win;
  const int oA = (ot << 4) + rowLo;                       // A row: out channel
  const float* wbase = w + ((size_t)oA * icPerG + kSel) * KS2;
  const int iwn = ((wt << 4) + rowLo) * stride - pad;     // B col pixel (pre-kx)

  v8f acc = {0.f, 0.f, 0.f, 0.f, 0.f, 0.f, 0.f, 0.f};
  #pragma unroll
  for (int ky = 0; ky < KS; ++ky) {
    int ih = ho * stride + ky - pad;
    if ((unsigned)ih >= (unsigned)Hin) continue;          // wave-uniform
    #pragma unroll
    for (int kx = 0; kx < KS; ++kx) {
      const int tap = ky * KS + kx;
      const int iw = iwn + kx;
      const bool ok = (unsigned)iw < (unsigned)Win;
      const float* ibase = in + ((size_t)b * Cin + icb + kSel) * chStride
                              + (size_t)ih * Win + (ok ? iw : 0);
      const float* wtap = wbase + tap;
      for (int k0 = 0; k0 < icPerG; k0 += 4) {
        __builtin_prefetch(ibase + (size_t)(k0 + 4) * chStride, 0, 0);
        v2f a, bb;
        a.x = wtap[(size_t)k0 * KS2];
        a.y = wtap[(size_t)(k0 + 1) * KS2];
        float bx = ibase[(size_t)k0 * chStride];
        float by = ibase[(size_t)(k0 + 1) * chStride];
        bb.x = ok ? bx : 0.f;
        bb.y = ok ? by : 0.f;
        acc = __builtin_amdgcn_wmma_f32_16x16x4_f32(
            false, a, false, bb, (short)0, acc, false, false);
      }
    }
  }
  const int mBase = (ot << 4) + ((lane >> 4) << 3);
  const int col = (wt << 4) + rowLo;
  #pragma unroll
  for (int i = 0; i < 8; ++i) {
    float v = acc[i];
    const int oc = mBase + i;
    if (EPI >= 1) v += bias[oc];
    if (EPI == 2) v = siluf(v);
    out[(((size_t)b * Cout + oc) * Hout + ho) * Wout + col] = v;
  }
}

// ---------------------------------------------------------------------------
// LayerNorm 1: gather token (s,q,d=0..63) from emb NCHW, LN over d -> tok_ln.
// ---------------------------------------------------------------------------
__global__ void ln1_kernel(const float* __restrict__ emb, const float* __restrict__ g,
                           const float* __restrict__ b, float* __restrict__ tok) {
  int token = blockIdx.x;                 // s*12 + q
  int d = threadIdx.x;                    // 0..63
  int s = token / 12, q = token - s * 12;
  int bb = s >> 10, hw = s & 1023;
  float v = emb[(((size_t)bb * 768 + q * 64 + d) << 10) + hw];
  __shared__ float red[64];
  red[d] = v; __syncthreads();
  for (int off = 32; off > 0; off >>= 1) { if (d < off) red[d] += red[d + off]; __syncthreads(); }
  float mean = red[0] * (1.f / 64.f);
  __syncthreads();
  float dv = v - mean;
  red[d] = dv * dv; __syncthreads();
  for (int off = 32; off > 0; off >>= 1) { if (d < off) red[d] += red[d + off]; __syncthreads(); }
  float var = red[0] * (1.f / 64.f);
  tok[(size_t)token * 64 + d] = dv * rsqrtf(var + 1e-5f) * g[d] + b[d];
}

// ---------------------------------------------------------------------------
// WMMA fp32 GEMM:  C[r][n] = sum_k A[r][k] * W[n][k]   (i.e. X @ W^T)
// A-row addressing: base = (r>>2)*aS + (r&3)*aT + aOff   (plain and chunked).
// EPI: 0 = none, 1 = +bias then softplus (dt projection)
// ---------------------------------------------------------------------------
template <int EPI>
__global__ void __launch_bounds__(128)
gemm_wmma_kernel(const float* __restrict__ A, const float* __restrict__ W,
                 const float* __restrict__ bias, float* __restrict__ C,
                 int K, int aS, int aT, int aOff, int ldc) {
  const int lane  = threadIdx.x & 31;
  const int wave  = threadIdx.x >> 5;
  const int rowLo = lane & 15;
  const int kSel  = (lane >> 4) << 1;
  const int tileN = blockIdx.y * 4 + wave;
  const int r = blockIdx.x * 16 + rowLo;
  const float* arow = A + (size_t)(r >> 2) * aS + (size_t)(r & 3) * aT + aOff;
  const int n = tileN * 16 + rowLo;
  const float* wrow = W + (size_t)n * K;
  v8f acc = {0.f, 0.f, 0.f, 0.f, 0.f, 0.f, 0.f, 0.f};
  for (int k0 = 0; k0 < K; k0 += 4) {
    __builtin_prefetch(arow + k0 + 32, 0, 0);   // global_prefetch_b8
    __builtin_prefetch(wrow + k0 + 32, 0, 0);
    v2f a, b;
    a.x = arow[k0 + kSel];
    a.y = arow[k0 + kSel + 1];
    b.x = wrow[k0 + kSel];
    b.y = wrow[k0 + kSel + 1];
    acc = __builtin_amdgcn_wmma_f32_16x16x4_f32(
        false, a, false, b, (short)0, acc, false, false);
  }
  const int mBase = blockIdx.x * 16 + ((lane >> 4) << 3);
  const int col = tileN * 16 + rowLo;
  #pragma unroll
  for (int i = 0; i < 8; ++i) {
    float v = acc[i];
    if (EPI == 1) {
      v += bias[col];
      v = (v > 20.f) ? v : log1pf(expf(v));   // softplus
    }
    C[(size_t)(mBase + i) * ldc + col] = v;
  }
}

// xi = silu(xz[:, :128] * conv_w + conv_b) in place (z = xz[:,128:] untouched)
__global__ void xi_silu_kernel(float* __restrict__ xz, const float* __restrict__ cw,
                               const float* __restrict__ cb) {
  int idx = blockIdx.x * 256 + threadIdx.x;   // total 40960*128
  int r = idx >> 7, c = idx & 127;
  float* p = xz + (size_t)r * 256 + c;
  float v = (*p) * cw[c] + cb[c];
  *p = siluf(v);
}

// ---------------------------------------------------------------------------
// Selective scan over one chunk of 4 tokens. Thread = (sequence s, channel d),
// keeps st[16] in registers, persists state via global buffer across chunks.
// STORE_Y: also emit y = (st.C + D*x) * silu(z), written in place over dt.
// ---------------------------------------------------------------------------
template <int STORE_Y>
__global__ void scan_kernel(const float* __restrict__ xz, const float* __restrict__ xdb,
                            const float* __restrict__ dt, const float* __restrict__ A_log,
                            const float* __restrict__ Dp, float* __restrict__ state,
                            float* __restrict__ yout) {
  int idx = blockIdx.x * 256 + threadIdx.x;   // total 10240*128
  int s = idx >> 7, d = idx & 127;
  float a[16];
  #pragma unroll
  for (int n = 0; n < 16; ++n) a[n] = -expf(A_log[d * 16 + n]);
  float st[16];
  float* sp = state + (size_t)idx * 16;
  #pragma unroll
  for (int n = 0; n < 16; ++n) st[n] = sp[n];
  float Dd = Dp[d];
  #pragma unroll
  for (int t = 0; t < 4; ++t) {
    int rt = s * 4 + t;
    float dtv = dt[(size_t)rt * 128 + d];
    float xv  = xz[(size_t)rt * 256 + d];
    const float* Bv = xdb + (size_t)rt * 64 + 32;
    const float* Cv = Bv + 16;
    float dx = dtv * xv;
    float y = 0.f;
    #pragma unroll
    for (int n = 0; n < 16; ++n) {
      st[n] = st[n] * expf(dtv * a[n]) + dx * Bv[n];
      if (STORE_Y) y += st[n] * Cv[n];
    }
    if (STORE_Y) {
      y += Dd * xv;
      float z = xz[(size_t)rt * 256 + 128 + d];
      yout[(size_t)rt * 128 + d] = y * siluf(z);
    }
  }
  #pragma unroll
  for (int n = 0; n < 16; ++n) sp[n] = st[n];
}

// LayerNorm 2: LN(mfi + mamba_out) over d, write NCHW y2 (10,256,32,32).
__global__ void ln2_kernel(const float* __restrict__ emb, const float* __restrict__ mo,
                           const float* __restrict__ g, const float* __restrict__ b,
                           float* __restrict__ y2) {
  int token = blockIdx.x;              // s*4 + t, 40960 tokens
  int d = threadIdx.x;                 // 0..63
  int s = token >> 2, t = token & 3;
  int bb = s >> 10, hw = s & 1023;
  float v = emb[(((size_t)bb * 768 + (8 + t) * 64 + d) << 10) + hw] +
            mo[(size_t)token * 64 + d];
  __shared__ float red[64];
  red[d] = v; __syncthreads();
  for (int off = 32; off > 0; off >>= 1) { if (d < off) red[d] += red[d + off]; __syncthreads(); }
  float mean = red[0] * (1.f / 64.f);
  __syncthreads();
  float dv = v - mean;
  red[d] = dv * dv; __syncthreads();
  for (int off = 32; off > 0; off >>= 1) { if (d < off) red[d] += red[d + off]; __syncthreads(); }
  float var = red[0] * (1.f / 64.f);
  y2[(((size_t)bb * 256 + t * 64 + d) << 10) + hw] =
      dv * rsqrtf(var + 1e-5f) * g[d] + b[d];
}

// Branch fuse-conv input: ch 0..63 = silu(bn(nearest2x(t0))), ch 64..127 = fid
__global__ void build_fuse_in_kernel(const float* __restrict__ t0,
                                     const float* __restrict__ fid,
                                     const float* __restrict__ ss,
                                     float* __restrict__ dst) {
  int idx = blockIdx.x * 256 + threadIdx.x;   // total 10*128*4096
  int p = idx & 4095; int t = idx >> 12;
  int c = t & 127; int b = t >> 7;
  float v;
  if (c < 64) {
    int h = p >> 6, w = p & 63;
    v = t0[(((size_t)b * 64 + c) << 10) + (h >> 1) * 32 + (w >> 1)];
    v = siluf(v * ss[2 * c] + ss[2 * c + 1]);
  } else {
    v = fid[(((size_t)b * 64 + (c - 64)) << 12) + p];
  }
  dst[idx] = v;
}

// final_j = silu(bn(fus_raw)) written into channel-concat buffer at chOff
__global__ void bn_apply_silu_cat_kernel(const float* __restrict__ x,
                                         const float* __restrict__ ss,
                                         float* __restrict__ dst, int chOff) {
  int idx = blockIdx.x * 256 + threadIdx.x;   // total 10*64*4096
  int p = idx & 4095; int t = idx >> 12;
  int c = t & 63; int b = t >> 6;
  float v = siluf(x[idx] * ss[2 * c] + ss[2 * c + 1]);
  dst[(((size_t)b * 128 + chOff + c) << 12) + p] = v;
}

// ---------------------------------------------------------------------------
// Host orchestration
// ---------------------------------------------------------------------------
extern "C" void kernel_launch(void* const* d_in, const int* in_sizes, int n_in,
                              void* d_out, int out_size, void* d_ws, size_t ws_size,
                              hipStream_t stream) {
  (void)in_sizes; (void)n_in; (void)out_size; (void)ws_size;
  const float* features = (const float*)d_in[0];
  const float* high     = (const float*)d_in[1];
  const float* fid0     = (const float*)d_in[2];
  const float* fid1     = (const float*)d_in[3];
  const float* bn_in_g  = (const float*)d_in[4];
  const float* bn_in_b  = (const float*)d_in[5];
  const float* down_w   = (const float*)d_in[6];
  const float* down_bn_g= (const float*)d_in[7];
  const float* down_bn_b= (const float*)d_in[8];
  const float* emb_w    = (const float*)d_in[9];
  const float* emb_b    = (const float*)d_in[10];
  const float* ln1_g    = (const float*)d_in[11];
  const float* ln1_b    = (const float*)d_in[12];
  const float* m_in_w   = (const float*)d_in[13];
  const float* m_conv_w = (const float*)d_in[14];
  const float* m_conv_b = (const float*)d_in[15];
  const float* m_xproj_w= (const float*)d_in[16];
  const float* m_dt_w   = (const float*)d_in[17];
  const float* m_dt_b   = (const float*)d_in[18];
  const float* m_A_log  = (const float*)d_in[19];
  const float* m_D      = (const float*)d_in[20];
  const float* m_out_w  = (const float*)d_in[21];
  const float* ln2_g    = (const float*)d_in[22];
  const float* ln2_b    = (const float*)d_in[23];
  const float* dec_w    = (const float*)d_in[24];
  const float* dec_b    = (const float*)d_in[25];
  const float* up_w_[2]    = { (const float*)d_in[26], (const float*)d_in[29] };
  const float* up_bn_g_[2] = { (const float*)d_in[27], (const float*)d_in[30] };
  const float* up_bn_b_[2] = { (const float*)d_in[28], (const float*)d_in[31] };
  const float* fus_w_[2]   = { (const float*)d_in[32], (const float*)d_in[36] };
  const float* fus_b_[2]   = { (const float*)d_in[33], (const float*)d_in[37] };
  const float* fus_bn_g_[2]= { (const float*)d_in[34], (const float*)d_in[38] };
  const float* fus_bn_b_[2]= { (const float*)d_in[35], (const float*)d_in[39] };
  const float* outf_w   = (const float*)d_in[40];
  const float* outf_b   = (const float*)d_in[41];
  float* out = (float*)d_out;

  float* ws = (float*)d_ws;
  // workspace layout (floats)
  const size_t o_f   = 0;                       // f_bn        (10,240,64,64)  9,830,400
  const size_t o_dn  = o_f   +  9830400;        // down        (10,384,32,32)  3,932,160
  const size_t o_emb = o_dn  +  3932160;        // emb         (10,768,32,32)  7,864,320
  const size_t o_tok = o_emb +  7864320;        // tok_ln      (10240,12,64)   7,864,320
  const size_t o_st  = o_tok +  7864320;        // scan state  (10240,128,16) 20,971,520
  const size_t o_xz  = o_st  + 20971520;        // xz          (40960,256)    10,485,760
  const size_t o_xdb = o_xz  + 10485760;        // xdb         (40960,64)      2,621,440
  const size_t o_dt  = o_xdb +  2621440;        // dt / y      (40960,128)     5,242,880
  const size_t o_mo  = o_dt  +  5242880;        // mamba_out   (40960,64)      2,621,440
  const size_t o_ss  = o_mo  +  2621440;        // BN scale/shift scratch      4,096
  // post-mamba reuse of early regions:
  float* y2      = ws + o_f;                    // (10,256,32,32)  2,621,440
  float* dec_out = ws + o_f + 2621440;          // (10,128,32,32)  1,310,720
  float* t0      = ws + o_f + 3932160;          // (10,64,32,32)     655,360
  float* fuse_in = ws + o_dt;                   // (10,128,64,64)  5,242,880
  float* fus_raw = ws + o_xz;                   // (10,64,64,64)   2,621,440
  float* outcat  = ws + o_xz + 2621440;         // (10,128,64,64)  5,242,880
  float* ss      = ws + o_ss;
  const int SS_IN = 0, SS_DN = 480, SS_UP = 1248, SS_FUS = 1376;

  // ---- stage 1: input BN (stats over virtual concat, then materialize) ----
  bn_in_stats_kernel<<<240, 256, 0, stream>>>(features, high, bn_in_g, bn_in_b, ss + SS_IN);
  build_f_kernel<<<38400, 256, 0, stream>>>(features, high, ss + SS_IN, ws + o_f);

  // ---- stage 2: down conv (3x3 s2 g12, WMMA) + BN + silu ----
  // waves = 10 * (384/16) * 32 * (32/16) = 15360 -> 3840 blocks
  conv_wmma_kernel<3, 0><<<3840, 128, 0, stream>>>(
      ws + o_f, down_w, nullptr, ws + o_dn,
      10, 240, 384, 64, 64, 32, 32, 2, 20, 32, 0);
  bn_stats_kernel<<<384, 256, 0, stream>>>(ws + o_dn, 384, 1024, 10,
                                           down_bn_g, down_bn_b, ss + SS_DN);
  bn_apply_silu_kernel<<<15360, 256, 0, stream>>>(ws + o_dn, ss + SS_DN, 384, 1024, 3932160);

  // ---- stage 3: emb conv (3x3 s1 g12, bias, WMMA) ----
  // waves = 10 * 48 * 32 * 2 = 30720 -> 7680 blocks
  conv_wmma_kernel<3, 1><<<7680, 128, 0, stream>>>(
      ws + o_dn, emb_w, emb_b, ws + o_emb,
      10, 384, 768, 32, 32, 32, 32, 1, 32, 64, 0);

  // ---- stage 4: LayerNorm 1 -> tok_ln ----
  ln1_kernel<<<122880, 64, 0, stream>>>(ws + o_emb, ln1_g, ln1_b, ws + o_tok);

  // ---- stage 5: Mamba chunks (state carried through workspace) ----
  hipMemsetAsync(ws + o_st, 0, (size_t)20971520 * sizeof(float), stream);
  for (int i = 0; i < 3; ++i) {
    // G1: xz = tok_chunk @ in_w^T   (M=40960, N=256, K=64)
    gemm_wmma_kernel<0><<<dim3(2560, 4), 128, 0, stream>>>(
        ws + o_tok, m_in_w + i * 16384, nullptr, ws + o_xz,
        64, /*aS=*/768, /*aT=*/64, /*aOff=*/256 * i, /*ldc=*/256);
    // xi = silu(xi*conv_w + conv_b) in place
    xi_silu_kernel<<<20480, 256, 0, stream>>>(ws + o_xz, m_conv_w + i * 128, m_conv_b + i * 128);
    // G2: xdb = xi @ xproj_w^T      (N=64, K=128)
    gemm_wmma_kernel<0><<<dim3(2560, 1), 128, 0, stream>>>(
        ws + o_xz, m_xproj_w + i * 8192, nullptr, ws + o_xdb,
        128, 1024, 256, 0, 64);
    // G3: dt = softplus(xdb[:, :32] @ dt_w^T + dt_b)   (N=128, K=32)
    gemm_wmma_kernel<1><<<dim3(2560, 2), 128, 0, stream>>>(
        ws + o_xdb, m_dt_w + i * 4096, m_dt_b + i * 128, ws + o_dt,
        32, 256, 64, 0, 128);
    // scan (y only for last chunk; y overwrites dt in place, same index/thread)
    if (i < 2)
      scan_kernel<0><<<5120, 256, 0, stream>>>(ws + o_xz, ws + o_xdb, ws + o_dt,
                                               m_A_log + i * 2048, m_D + i * 128,
                                               ws + o_st, nullptr);
    else
      scan_kernel<1><<<5120, 256, 0, stream>>>(ws + o_xz, ws + o_xdb, ws + o_dt,
                                               m_A_log + i * 2048, m_D + i * 128,
                                               ws + o_st, ws + o_dt);
  }
  // G4: mamba_out = y @ out_w^T (chunk 2 weights)  (N=64, K=128)
  gemm_wmma_kernel<0><<<dim3(2560, 1), 128, 0, stream>>>(
      ws + o_dt, m_out_w + 2 * 8192, nullptr, ws + o_mo,
      128, 512, 128, 0, 64);

  // ---- stage 6: LayerNorm 2 + reshape to NCHW ----
  ln2_kernel<<<40960, 64, 0, stream>>>(ws + o_emb, ws + o_mo, ln2_g, ln2_b, y2);

  // ---- stage 7: dec conv (3x3 g4, bias, WMMA) ----
  // waves = 10 * 8 * 32 * 2 = 5120 -> 1280 blocks
  conv_wmma_kernel<3, 1><<<1280, 128, 0, stream>>>(
      y2, dec_w, dec_b, dec_out,
      10, 256, 128, 32, 32, 32, 32, 1, 64, 32, 0);

  // ---- stage 8: two upsample/fuse branches (all convs WMMA) ----
  for (int j = 0; j < 2; ++j) {
    // 1x1 up-conv on mid slice (equivalent to conv after nearest-up; BN stats of
    // 4x-repeated data equal stats of the 32x32 map). waves = 10*4*32*2 = 2560.
    conv_wmma_kernel<1, 0><<<640, 128, 0, stream>>>(
        dec_out, up_w_[j], nullptr, t0,
        10, 128, 64, 32, 32, 32, 32, 1, 64, 64, /*icOffset=*/j * 64);
    bn_stats_kernel<<<64, 256, 0, stream>>>(t0, 64, 1024, 10,
                                            up_bn_g_[j], up_bn_b_[j], ss + SS_UP);
    // fuse-conv input: concat(silu(bn(up2x(t0))), fid)
    build_fuse_in_kernel<<<20480, 256, 0, stream>>>(t0, j ? fid1 : fid0,
                                                    ss + SS_UP, fuse_in);
    // fuse conv 3x3 (128 -> 64, bias). waves = 10*4*64*4 = 10240 -> 2560 blocks
    conv_wmma_kernel<3, 1><<<2560, 128, 0, stream>>>(
        fuse_in, fus_w_[j], fus_b_[j], fus_raw,
        10, 128, 64, 64, 64, 64, 64, 1, 128, 64, 0);
    bn_stats_kernel<<<64, 256, 0, stream>>>(fus_raw, 64, 4096, 10,
                                            fus_bn_g_[j], fus_bn_b_[j], ss + SS_FUS);
    bn_apply_silu_cat_kernel<<<10240, 256, 0, stream>>>(fus_raw, ss + SS_FUS,
                                                        outcat, j * 64);
  }

  // ---- stage 9: final 1x1 conv + bias + silu -> d_out (WMMA) ----
  // waves = 10 * 4 * 64 * 4 = 10240 -> 2560 blocks
  conv_wmma_kernel<1, 2><<<2560, 128, 0, stream>>>(
      outcat, outf_w, outf_b, out,
      10, 128, 64, 64, 64, 64, 64, 1, 128, 64, 0);
}